// HybridAttentionLayer_1340029796636
// MI455X (gfx1250) — compile-verified
//
#include <hip/hip_runtime.h>
#include <hip/hip_bf16.h>
#include <math.h>

#define H   512
#define BSZ 16
#define LQd 64
#define LAd 100

typedef __attribute__((ext_vector_type(2))) float v2f;
typedef __attribute__((ext_vector_type(8))) float v8f;

__device__ __forceinline__ float fast_tanh(float x) {
#if __has_builtin(__builtin_amdgcn_tanhf)
    return __builtin_amdgcn_tanhf(x);
#elif __has_builtin(__builtin_amdgcn_tanh_f32)
    return __builtin_amdgcn_tanh_f32(x);
#else
    return tanhf(x);
#endif
}

// ---------------------------------------------------------------------------
// Kernel 1: a1[r,h] = c1 @ W_a + (c1*c1) @ W_aq + (b_a + b_aq)
// r in [0,1024) = flat (b,lq). One wave per 16x16 output tile, K=512 in steps
// of 4 via V_WMMA_F32_16X16X4_F32 (fp32-native WMMA, full precision).
// A-frag layout (ISA 7.12.2): lane<16 holds K={k,k+1}, lane>=16 K={k+2,k+3},
// row = lane&15.  C layout: vgpr i -> row i + 8*(lane>>4), col = lane&15.
// ---------------------------------------------------------------------------
__global__ __launch_bounds__(256) void proj_a_kernel(
    const float* __restrict__ c1, const float* __restrict__ Wa,
    const float* __restrict__ Waq, const float* __restrict__ ba,
    const float* __restrict__ baq, float* __restrict__ a1)
{
    const int gw   = (blockIdx.x * blockDim.x + threadIdx.x) >> 5; // global wave
    const int lane = threadIdx.x & 31;
    const int TN   = H / 16;              // 32 col tiles
    const int tm   = gw / TN;             // 0..63
    const int tn   = gw % TN;
    const int half = lane >> 4;
    const int lo   = lane & 15;
    const int arow = tm * 16 + lo;        // A-matrix row for this lane
    const int bcol = tn * 16 + lo;        // B/C column for this lane

    v8f acc = {};
    #pragma unroll 4
    for (int k = 0; k < H; k += 4) {
        const int ka = k + 2 * half;
        v2f a;  a.x  = c1[arow * H + ka];        a.y  = c1[arow * H + ka + 1];
        v2f a2 = a * a;
        v2f b0; b0.x = Wa [ka * H + bcol];       b0.y = Wa [(ka + 1) * H + bcol];
        v2f b1; b1.x = Waq[ka * H + bcol];       b1.y = Waq[(ka + 1) * H + bcol];
        acc = __builtin_amdgcn_wmma_f32_16x16x4_f32(false, a,  false, b0, (short)0, acc, false, false);
        acc = __builtin_amdgcn_wmma_f32_16x16x4_f32(false, a2, false, b1, (short)0, acc, false, false);
    }
    const float bias = ba[bcol] + baq[bcol];
    #pragma unroll
    for (int i = 0; i < 8; ++i) {
        const int r = tm * 16 + half * 8 + i;
        a1[r * H + bcol] = acc[i] + bias;
    }
}

// ---------------------------------------------------------------------------
// Kernel 2: q2[r,h] = c2 @ W_q + b_q,  r in [0,1600) = flat (b,la).
// ---------------------------------------------------------------------------
__global__ __launch_bounds__(256) void proj_q_kernel(
    const float* __restrict__ c2, const float* __restrict__ Wq,
    const float* __restrict__ bq, float* __restrict__ q2)
{
    const int gw   = (blockIdx.x * blockDim.x + threadIdx.x) >> 5;
    const int lane = threadIdx.x & 31;
    const int TN   = H / 16;              // 32
    const int tm   = gw / TN;             // 0..99
    const int tn   = gw % TN;
    const int half = lane >> 4;
    const int lo   = lane & 15;
    const int arow = tm * 16 + lo;
    const int bcol = tn * 16 + lo;

    v8f acc = {};
    #pragma unroll 4
    for (int k = 0; k < H; k += 4) {
        const int ka = k + 2 * half;
        v2f a;  a.x  = c2[arow * H + ka];  a.y  = c2[arow * H + ka + 1];
        v2f b0; b0.x = Wq[ka * H + bcol];  b0.y = Wq[(ka + 1) * H + bcol];
        acc = __builtin_amdgcn_wmma_f32_16x16x4_f32(false, a, false, b0, (short)0, acc, false, false);
    }
    const float bias = bq[bcol];
    #pragma unroll
    for (int i = 0; i < 8; ++i) {
        const int r = tm * 16 + half * 8 + i;
        q2[r * H + bcol] = acc[i] + bias;
    }
}

// ---------------------------------------------------------------------------
// Kernel 3 (the hot one, fused): one block per (b,lq).
//   score[la] = sum_h tanh(a1[b,lq,h] + q2[b,la,h]) * Wv[h]     (b_v dropped:
//   uniform shift is softmax/entropy-invariant)
//   beta = logsumexp(score) - sum(softmax(score)*score)
// a1 row + Wv staged in LDS; 8 waves split the 100 'la' rows; wave32 shfl
// reductions; one wave finishes the entropy over the 100 scores.
// ---------------------------------------------------------------------------
__global__ __launch_bounds__(256) void score_entropy_kernel(
    const float* __restrict__ a1, const float* __restrict__ q2,
    const float* __restrict__ Wv, float* __restrict__ out)
{
    __shared__ float sA[H];
    __shared__ float sW[H];
    __shared__ float sScore[LAd];

    const int tid  = threadIdx.x;
    const int b    = blockIdx.x / LQd;
    const int wave = tid >> 5;
    const int lane = tid & 31;

    // stage a1 row for this (b,lq) and Wv
    for (int i = tid; i < H; i += 256) {
        sA[i] = a1[blockIdx.x * H + i];
        sW[i] = Wv[i];
    }
    __syncthreads();

    const float* q2b = q2 + (size_t)b * LAd * H;
    for (int la = wave; la < LAd; la += 8) {
        const float* qr = q2b + la * H;
        float s = 0.0f;
        #pragma unroll
        for (int j = 0; j < H / 32; ++j) {
            const int h = lane + 32 * j;           // coalesced across the wave
            s = fmaf(fast_tanh(sA[h] + qr[h]), sW[h], s);
        }
        #pragma unroll
        for (int m = 16; m >= 1; m >>= 1)
            s += __shfl_xor(s, m, 32);
        if (lane == 0) sScore[la] = s;
    }
    __syncthreads();

    if (wave == 0) {
        // lane handles la = lane, lane+32, lane+64, (+96 if lane<4)
        const float NEG = -3.0e38f;
        float x0 = sScore[lane];
        float x1 = sScore[lane + 32];
        float x2 = sScore[lane + 64];
        float x3 = (lane < 4) ? sScore[lane + 96] : NEG;
        float mx = fmaxf(fmaxf(x0, x1), fmaxf(x2, x3));
        #pragma unroll
        for (int m = 16; m >= 1; m >>= 1)
            mx = fmaxf(mx, __shfl_xor(mx, m, 32));
        float e0 = __expf(x0 - mx), e1 = __expf(x1 - mx), e2 = __expf(x2 - mx);
        float e3 = (lane < 4) ? __expf(x3 - mx) : 0.0f;
        float se = e0 + e1 + e2 + e3;
        float ss = e0 * x0 + e1 * x1 + e2 * x2 + ((lane < 4) ? e3 * x3 : 0.0f);
        #pragma unroll
        for (int m = 16; m >= 1; m >>= 1) {
            se += __shfl_xor(se, m, 32);
            ss += __shfl_xor(ss, m, 32);
        }
        if (lane == 0) {
            const float L = mx + __logf(se);       // logsumexp
            out[blockIdx.x] = L - ss / se;         // entropy
        }
    }
}

// ---------------------------------------------------------------------------
// Host launcher. Input dict order:
// 0:content1 1:content2 2:is_user 3:W_q 4:b_q 5:W_a 6:b_a 7:W_aq 8:b_aq
// 9:W_v 10:b_v   (is_user==0 branch only, as in the reference; b_v unused)
// ---------------------------------------------------------------------------
extern "C" void kernel_launch(void* const* d_in, const int* in_sizes, int n_in,
                              void* d_out, int out_size, void* d_ws, size_t ws_size,
                              hipStream_t stream) {
    const float* c1  = (const float*)d_in[0];
    const float* c2  = (const float*)d_in[1];
    const float* Wq  = (const float*)d_in[3];
    const float* bq  = (const float*)d_in[4];
    const float* Wa  = (const float*)d_in[5];
    const float* ba  = (const float*)d_in[6];
    const float* Waq = (const float*)d_in[7];
    const float* baq = (const float*)d_in[8];
    const float* Wv  = (const float*)d_in[9];
    float* out = (float*)d_out;

    float* a1 = (float*)d_ws;                                   // 1024*512 f32 = 2 MB
    float* q2 = (float*)((char*)d_ws + (size_t)BSZ * LQd * H * sizeof(float));

    // (B*LQ/16)*(H/16) = 64*32 = 2048 tiles, 1 wave each, 8 waves/block
    proj_a_kernel<<<256, 256, 0, stream>>>(c1, Wa, Waq, ba, baq, a1);
    // (B*LA/16)*(H/16) = 100*32 = 3200 tiles
    proj_q_kernel<<<400, 256, 0, stream>>>(c2, Wq, bq, q2);
    // one block per (b,lq)
    score_entropy_kernel<<<BSZ * LQd, 256, 0, stream>>>(a1, q2, Wv, out);
}